// VocabParallelEmbeddingWithPacked_28973849379116
// MI455X (gfx1250) — compile-verified
//
#include <hip/hip_runtime.h>

// 16-byte vector (8 x fp16) for B128 global loads/stores.
typedef __attribute__((ext_vector_type(4))) unsigned int v4u;

// One workgroup per token. Copies one D-element fp16 row from the gathered
// adapter table to the output, or zeros it if the token is outside this TP
// shard's vocab range. All bulk traffic is 128-bit, non-temporal.
__global__ __launch_bounds__(256) void vocab_packed_embed_gather(
    const int* __restrict__ x,            // [N] token ids
    const v4u* __restrict__ weights,      // [P, V, D] fp16 viewed as 16B vectors
    const int* __restrict__ indices,      // [N] adapter index
    const int* __restrict__ vocab_start,  // scalar (device)
    const int* __restrict__ vocab_end,    // scalar (device)
    v4u* __restrict__ out,                // [N, D] fp16 viewed as 16B vectors
    int nvec_per_row)                     // D / 8
{
    const int n   = blockIdx.x;
    const int vs  = vocab_start[0];
    const int ve  = vocab_end[0];
    const int tok = x[n];                 // wave-uniform -> s_load

    v4u* __restrict__ orow = out + (long long)n * nvec_per_row;

    const bool oob = (tok < vs) | (tok >= ve);
    if (oob) {
        // Out-of-shard token: row is zero after the (identity) all-reduce.
        const v4u z = {0u, 0u, 0u, 0u};
        for (int i = threadIdx.x; i < nvec_per_row; i += blockDim.x) {
            __builtin_nontemporal_store(z, &orow[i]);
        }
        return;
    }

    // Per-shard vocab size V = ve - vs; adapter table stride = V * D elements.
    const long long table_stride_vec = (long long)(ve - vs) * (long long)nvec_per_row;
    const long long row              = (long long)(tok - vs) * (long long)nvec_per_row;
    const v4u* __restrict__ wrow =
        weights + (long long)indices[n] * table_stride_vec + row;

    // D=4096 -> 512 vectors per row; 256 threads -> 2 vectors per thread.
    // Unroll so both B128 loads are in flight before the stores retire.
#pragma unroll 4
    for (int i = threadIdx.x; i < nvec_per_row; i += blockDim.x) {
        v4u v = __builtin_nontemporal_load(&wrow[i]);
        __builtin_nontemporal_store(v, &orow[i]);
    }
}

extern "C" void kernel_launch(void* const* d_in, const int* in_sizes, int n_in,
                              void* d_out, int out_size, void* d_ws, size_t ws_size,
                              hipStream_t stream) {
    (void)d_ws; (void)ws_size; (void)n_in;

    const int* x        = (const int*)d_in[0];
    const v4u* weights  = (const v4u*)d_in[1];
    const int* indices  = (const int*)d_in[2];
    const int* vs       = (const int*)d_in[3];  // single-element device array
    const int* ve       = (const int*)d_in[4];  // single-element device array

    const int N    = in_sizes[0];        // tokens
    const int D    = out_size / N;       // embed dim (4096)
    const int nvec = D / 8;              // fp16 elements per 16B vector

    vocab_packed_embed_gather<<<N, 256, 0, stream>>>(
        x, weights, indices, vs, ve, (v4u*)d_out, nvec);
}